// GCN_69088843923992
// MI455X (gfx1250) — compile-verified
//
#include <hip/hip_runtime.h>

typedef float v2f __attribute__((ext_vector_type(2)));
typedef float v8f __attribute__((ext_vector_type(8)));
typedef int   vi4 __attribute__((vector_size(16)));   // matches builtin param type

#define N_NODES 50000
#define N_EDGES 800000
#define F_IN    128
#define F_HID   256
#define F_OUT   64
#define REPS    24   // batch_size(2) * pred_steps(12)

#define LDS_STRIDE 80   // 64 data floats + 16 pad -> half-waves hit disjoint banks

// ---- async global->LDS helpers ----
__device__ __forceinline__ void async_copy_b128(const float* g, float* l) {
#if __has_builtin(__builtin_amdgcn_global_load_async_to_lds_b128)
    __builtin_amdgcn_global_load_async_to_lds_b128(
        (__attribute__((address_space(1))) vi4*)g,     // global src (AS1)
        (__attribute__((address_space(3))) vi4*)l,     // LDS dst (AS3)
        /*imm offset*/0, /*imm cpol*/0);
#else
    *(float4*)l = *(const float4*)g;     // fallback: load + ds_store
#endif
}

__device__ __forceinline__ void async_wait0() {
#if __has_builtin(__builtin_amdgcn_s_wait_asynccnt)
    __builtin_amdgcn_s_wait_asynccnt(0);
#else
    asm volatile("s_wait_asynccnt 0" ::: "memory");
#endif
}

// ---------------- degree / normalization ----------------
__global__ void k_init_deg(float* __restrict__ deg) {
    int i = blockIdx.x * blockDim.x + threadIdx.x;
    if (i < N_NODES) deg[i] = 1.0f;          // self-loop
}

__global__ void k_deg_scatter(const int* __restrict__ row, float* __restrict__ deg) {
    int e = blockIdx.x * blockDim.x + threadIdx.x;
    if (e < N_EDGES) atomicAdd(&deg[row[e]], 1.0f);
}

__global__ void k_deg_to_dinv(float* __restrict__ deg) {
    int i = blockIdx.x * blockDim.x + threadIdx.x;
    if (i < N_NODES) deg[i] = rsqrtf(deg[i]);   // deg >= 1 always
}

// ---------------- dense GEMM via V_WMMA_F32_16X16X4_F32 ----------------
// Block = 128 threads = 4 waves; the block owns a 16-row M-tile and a 64-col
// N-panel. The K x 64 weight panel is staged into LDS once via the async
// global->LDS engine (ASYNCcnt), then each wave streams 16x16 WMMA tiles,
// reading B fragments from LDS (ds_load_b32, bank-conflict-free via padding).
// A fragment (16x4 f32): lane 0-15 -> M=lane, regs {K0,K1}; lane 16-31 -> {K2,K3}.
// B fragment (4x16 f32): lane 0-15 -> N=lane, regs {K0,K1}; lane 16-31 -> {K2,K3}.
// C/D (16x16 f32): lane 0-15 -> N=lane, vgpr v -> M=v; lane 16-31 -> M=v+8.
template <int K, int NOUT>
__global__ void k_gemm_wmma(const float* __restrict__ A,
                            const float* __restrict__ W,
                            float* __restrict__ Hout) {
    __shared__ float ldsW[K * LDS_STRIDE];

    const int c0 = blockIdx.y * 64;          // first column of this block's panel
    // stage K x 64 panel: each step copies 16B; 16 segments of 4 floats per row
    for (int i = threadIdx.x; i < K * 16; i += 128) {
        int krow = i >> 4;
        int seg  = i & 15;
        async_copy_b128(W + (size_t)krow * NOUT + c0 + seg * 4,
                        &ldsW[krow * LDS_STRIDE + seg * 4]);
    }
    async_wait0();
    __syncthreads();

    const int lane = threadIdx.x & 31;
    const int wave = threadIdx.x >> 5;
    const int mt   = blockIdx.x;
    const int r    = lane & 15;
    const int hi   = lane >> 4;
    const int m    = mt * 16 + r;            // A row for this lane
    const int wn   = wave * 16 + r;          // column within the 64-wide panel
    const int n    = c0 + wn;                // global output column

    v8f c = {};
    const float* Arow = A + (size_t)m * K;
#pragma unroll 4
    for (int kc = 0; kc < K / 4; ++kc) {
        const int k0 = kc * 4 + hi * 2;
        v2f a = *(const v2f*)(Arow + k0);
        v2f b;
        b.x = ldsW[(k0 + 0) * LDS_STRIDE + wn];
        b.y = ldsW[(k0 + 1) * LDS_STRIDE + wn];
        c = __builtin_amdgcn_wmma_f32_16x16x4_f32(
                /*neg_a=*/false, a, /*neg_b=*/false, b,
                /*c_mod=*/(short)0, c, /*reuse_a=*/false, /*reuse_b=*/false);
    }
#pragma unroll
    for (int v = 0; v < 8; ++v) {
        Hout[(size_t)(mt * 16 + hi * 8 + v) * NOUT + n] = c[v];
    }
}

// ---------------- aggregation ----------------
// agg[i,:] = h[i,:] * dinv[i]^2 + bias   (self-loop term + bias)
template <int D>
__global__ void k_agg_init(const float* __restrict__ h,
                           const float* __restrict__ dinv,
                           const float* __restrict__ bias,
                           float* __restrict__ agg) {
    int idx = blockIdx.x * blockDim.x + threadIdx.x;
    if (idx >= N_NODES * D) return;
    int i = idx / D;
    int d = idx - i * D;
    float s = dinv[i];
    agg[idx] = h[idx] * (s * s) + bias[d];
}

// agg[row,:] += h[col,:] * dinv[row]*dinv[col]  (D/4 threads per edge, float4 gather)
template <int D>
__global__ void k_edge_scatter(const float* __restrict__ h,
                               const int* __restrict__ row,
                               const int* __restrict__ col,
                               const float* __restrict__ dinv,
                               float* __restrict__ agg) {
    const int TPE = D / 4;
    long long t = (long long)blockIdx.x * blockDim.x + threadIdx.x;
    int e = (int)(t / TPE);
    if (e >= N_EDGES) return;
    int f = (int)(t % TPE) * 4;
    int ri = row[e];
    int ci = col[e];
    float norm = dinv[ri] * dinv[ci];
    float4 v = *(const float4*)(h + (size_t)ci * D + f);
    float* dst = agg + (size_t)ri * D + f;
    atomicAdd(dst + 0, v.x * norm);
    atomicAdd(dst + 1, v.y * norm);
    atomicAdd(dst + 2, v.z * norm);
    atomicAdd(dst + 3, v.w * norm);
}

__global__ void k_relu(float* __restrict__ x, int n) {
    int i = blockIdx.x * blockDim.x + threadIdx.x;
    if (i < n) x[i] = fmaxf(x[i], 0.0f);
}

// ---------------- log_softmax over 64 + broadcast to [2,12,N,64] ----------------
// One wave32 per node; each lane owns 2 consecutive features.
__global__ void k_logsoftmax_bcast(const float* __restrict__ h,
                                   float* __restrict__ out) {
    int wave = threadIdx.x >> 5;
    int lane = threadIdx.x & 31;
    int node = blockIdx.x * (blockDim.x >> 5) + wave;
    if (node >= N_NODES) return;

    float2 x = *(const float2*)(h + (size_t)node * F_OUT + lane * 2);

    float m = fmaxf(x.x, x.y);
#pragma unroll
    for (int off = 16; off > 0; off >>= 1) m = fmaxf(m, __shfl_xor(m, off, 32));

    float s = expf(x.x - m) + expf(x.y - m);
#pragma unroll
    for (int off = 16; off > 0; off >>= 1) s += __shfl_xor(s, off, 32);

    float lse = m + logf(s);
    float2 o;
    o.x = x.x - lse;
    o.y = x.y - lse;

    size_t base = (size_t)node * F_OUT + lane * 2;
#pragma unroll
    for (int rep = 0; rep < REPS; ++rep) {
        *(float2*)(out + (size_t)rep * N_NODES * F_OUT + base) = o;
    }
}

// ---------------- launch ----------------
extern "C" void kernel_launch(void* const* d_in, const int* in_sizes, int n_in,
                              void* d_out, int out_size, void* d_ws, size_t ws_size,
                              hipStream_t stream) {
    (void)in_sizes; (void)n_in; (void)out_size; (void)ws_size;
    const float* x  = (const float*)d_in[0];
    const float* W1 = (const float*)d_in[1];
    const float* b1 = (const float*)d_in[2];
    const float* W2 = (const float*)d_in[3];
    const float* b2 = (const float*)d_in[4];
    const int*   ei = (const int*)d_in[5];
    const int* row = ei;            // edge_index[0]
    const int* col = ei + N_EDGES;  // edge_index[1]
    float* out = (float*)d_out;

    // workspace layout (floats), 256-float aligned sections
    float* ws   = (float*)d_ws;
    float* deg  = ws;                                   // N        (-> dinv in place)
    float* h1   = ws + 50176;                           // N*256
    float* agg1 = h1 + (size_t)N_NODES * F_HID;         // N*256
    float* h2   = agg1 + (size_t)N_NODES * F_HID;       // N*64
    float* agg2 = h2 + (size_t)N_NODES * F_OUT;         // N*64

    // normalization
    k_init_deg<<<(N_NODES + 255) / 256, 256, 0, stream>>>(deg);
    k_deg_scatter<<<(N_EDGES + 255) / 256, 256, 0, stream>>>(row, deg);
    k_deg_to_dinv<<<(N_NODES + 255) / 256, 256, 0, stream>>>(deg);

    // layer 1: h1 = x @ W1 ; agg1 = D^-1/2 (A+I) D^-1/2 h1 + b1 ; relu
    dim3 g1(N_NODES / 16, F_HID / 64);         // 3125 x 4 blocks, 4 waves/block
    k_gemm_wmma<F_IN, F_HID><<<g1, 128, 0, stream>>>(x, W1, h1);
    k_agg_init<F_HID><<<(N_NODES * F_HID + 255) / 256, 256, 0, stream>>>(h1, deg, b1, agg1);
    k_edge_scatter<F_HID><<<(int)(((long long)N_EDGES * (F_HID / 4) + 255) / 256), 256, 0, stream>>>(
        h1, row, col, deg, agg1);
    k_relu<<<(N_NODES * F_HID + 255) / 256, 256, 0, stream>>>(agg1, N_NODES * F_HID);

    // layer 2: h2 = relu(agg1) @ W2 ; agg2 = norm-agg + b2
    dim3 g2(N_NODES / 16, F_OUT / 64);         // 3125 x 1
    k_gemm_wmma<F_HID, F_OUT><<<g2, 128, 0, stream>>>(agg1, W2, h2);
    k_agg_init<F_OUT><<<(N_NODES * F_OUT + 255) / 256, 256, 0, stream>>>(h2, deg, b2, agg2);
    k_edge_scatter<F_OUT><<<(int)(((long long)N_EDGES * (F_OUT / 4) + 255) / 256), 256, 0, stream>>>(
        h2, row, col, deg, agg2);

    // log_softmax + broadcast to [2,12,N,64]
    k_logsoftmax_bcast<<<(N_NODES + 7) / 8, 256, 0, stream>>>(agg2, out);
}